// TxtNet_2611340116407
// MI455X (gfx1250) — compile-verified
//
#include <hip/hip_runtime.h>
#include <hip/hip_bf16.h>
#include <math.h>

typedef __attribute__((ext_vector_type(2))) float v2f;
typedef __attribute__((ext_vector_type(4))) float v4f;
typedef __attribute__((ext_vector_type(8))) float v8f;
typedef __attribute__((ext_vector_type(2))) int   v2i;
typedef __attribute__((ext_vector_type(4))) int   v4i;

#define NN   8192
#define EE   262144
#define FEAT 1386
#define HID  4096
#define CODE 64

#define TM 256   // block tile rows (8 waves x 2 stripes x 16)
#define TN 64    // block tile cols (4 wmma subtiles x 16)
#define KC 32    // K chunk staged in LDS (8 wmma k-steps, 64 WMMAs/wave/chunk)
#define APAD 4   // A row stride 36 floats: 8B-aligned b64 frags, conflict-free
#define BPAD 2   // B row stride 34 floats: 8B-aligned b64 frags

// ---- gfx1250 async global->LDS copy (ASYNCcnt path), with safe fallback ----
#if defined(__has_builtin)
#  if __has_builtin(__builtin_amdgcn_global_load_async_to_lds_b64)
#    define ASYNC_B64 1
#  endif
#endif

__device__ __forceinline__ void cp_b64(float* l, const float* g) {
#ifdef ASYNC_B64
    __builtin_amdgcn_global_load_async_to_lds_b64(
        (__attribute__((address_space(1))) v2i*)(void*)g,
        (__attribute__((address_space(3))) v2i*)(void*)l, 0, 0);
#else
    *(v2f*)l = *(const v2f*)g;
#endif
}

__device__ __forceinline__ void wait_async() {
#ifdef ASYNC_B64
#  if defined(__has_builtin) && __has_builtin(__builtin_amdgcn_s_wait_asynccnt)
    __builtin_amdgcn_s_wait_asynccnt(0);
#  else
    asm volatile("s_wait_asynccnt 0" ::: "memory");
#  endif
#endif
}

__device__ __forceinline__ v8f wmma4(v2f a, v2f b, v8f c) {
    return __builtin_amdgcn_wmma_f32_16x16x4_f32(
        /*neg_a=*/false, a, /*neg_b=*/false, b,
        /*c_mod=*/(short)0, c, /*reuse_a=*/false, /*reuse_b=*/false);
}

// ---------------------------------------------------------------------------
// fp32 GEMM: C[M,N] = op(A[M,K] @ B[K,N] + bias) via V_WMMA_F32_16X16X4_F32
// grid = (M/TM, N/TN), block = 256 threads (8 wave32)
// A: async global->LDS double buffer; B: register-pipelined transpose staging
// ---------------------------------------------------------------------------
__global__ __launch_bounds__(256)
void gemm_f32_wmma(const float* __restrict__ A, const float* __restrict__ B,
                   const float* __restrict__ bias, float* __restrict__ C,
                   int M, int K, int N, int do_relu)
{
    __shared__ float As[2][TM][KC + APAD];   // 2 x 36 KB
    __shared__ float Bs[2][TN][KC + BPAD];   // 2 x 8.5 KB, transposed: Bs[n][k]

    const int tid   = threadIdx.x;
    const int wave  = tid >> 5;
    const int lane  = tid & 31;
    const int lhalf = lane >> 4;
    const int lmod  = lane & 15;
    const int tileM = blockIdx.x * TM;
    const int tileN = blockIdx.y * TN;

    v8f acc[2][4];
    const v8f vzero = {0.f,0.f,0.f,0.f,0.f,0.f,0.f,0.f};
#pragma unroll
    for (int s = 0; s < 2; ++s)
#pragma unroll
        for (int n = 0; n < 4; ++n) acc[s][n] = vzero;

    const int nchunks = (K + KC - 1) / KC;
    const bool hastail = (K % KC) != 0;

    // --- A staging: 16 async b64 per thread (full chunks) ---
    auto stage_a = [&](int c, int buf) {
        int kk = c * KC;
        if (hastail && c == nchunks - 1) {
#pragma unroll
            for (int i = 0; i < (TM * KC) / 256; ++i) {      // scalar, zero-padded
                int idx = tid + i * 256;
                int r = idx >> 5, cc = idx & 31;
                int gc = kk + cc;
                As[buf][r][cc] = (gc < K) ? A[(size_t)(tileM + r) * K + gc] : 0.f;
            }
        } else {
#pragma unroll
            for (int i = 0; i < (TM * KC) / (256 * 2); ++i) { // 16 x b64 async
                int fidx = tid + i * 256;                     // 0..4095
                int r  = fidx >> 4;                           // 0..255
                int kg = (fidx & 15) * 2;                     // 0,2,..,30
                cp_b64(&As[buf][r][kg], &A[(size_t)(tileM + r) * K + kk + kg]);
            }
        }
    };

    // --- B staging: global -> regs (issued pre-compute) ---
    auto load_b = [&](int c, v4f* br) {
        int kk = c * KC;
        bool tail = hastail && c == nchunks - 1;
#pragma unroll
        for (int j = 0; j < 2; ++j) {
            int idx = tid + j * 256;                          // 0..511
            int bk = idx >> 4;                                // 0..31
            int bn = (idx & 15) * 4;                          // 0..60
            if (tail && (kk + bk) >= K) {
                v4f z = {0.f,0.f,0.f,0.f};
                br[j] = z;
            } else {
                br[j] = *(const v4f*)&B[(size_t)(kk + bk) * N + tileN + bn];
            }
        }
    };
    // --- B staging: regs -> LDS transposed (issued post-compute) ---
    auto store_b = [&](int buf, const v4f* br) {
#pragma unroll
        for (int j = 0; j < 2; ++j) {
            int idx = tid + j * 256;
            int bk = idx >> 4;
            int bn = (idx & 15) * 4;
#pragma unroll
            for (int u = 0; u < 4; ++u)
                Bs[buf][bn + u][bk] = br[j][u];
        }
    };

    // --- compute one staged chunk: 64 WMMAs per wave ---
    auto compute = [&](int buf) {
#pragma unroll
        for (int ks = 0; ks < KC; ks += 4) {
            int karow = ks + 2 * lhalf;
            v2f a0 = *(const v2f*)&As[buf][16 * wave + lmod][karow];
            v2f a1 = *(const v2f*)&As[buf][128 + 16 * wave + lmod][karow];
#pragma unroll
            for (int n = 0; n < 4; ++n) {
                v2f b = *(const v2f*)&Bs[buf][16 * n + lmod][karow];
                acc[0][n] = wmma4(a0, b, acc[0][n]);
                acc[1][n] = wmma4(a1, b, acc[1][n]);
            }
        }
    };

    // --- prologue: stage chunk 0 ---
    stage_a(0, 0);
    {
        v4f br[2];
        load_b(0, br);
        store_b(0, br);
    }

    // --- pipelined main loop ---
    int cur = 0;
    for (int c = 0; c < nchunks; ++c) {
        wait_async();
        __syncthreads();
        const bool pre = (c + 1 < nchunks);
        v4f br[2];
        if (pre) {
            stage_a(c + 1, cur ^ 1);   // async A -> LDS (other buffer)
            load_b(c + 1, br);         // B global -> regs (latency hidden by WMMAs)
        }
        compute(cur);
        if (pre) store_b(cur ^ 1, br); // B regs -> LDS transposed
        cur ^= 1;
    }

    // --- epilogue: C/D layout lane(l&15)=col, vgpr r -> row r + 8*(l>>4) ---
#pragma unroll
    for (int s = 0; s < 2; ++s)
#pragma unroll
        for (int n = 0; n < 4; ++n) {
            int gcol = tileN + 16 * n + lmod;
            float bv = bias ? bias[gcol] : 0.f;
#pragma unroll
            for (int r = 0; r < 8; ++r) {
                int grow = tileM + 128 * s + 16 * wave + r + 8 * lhalf;
                float v = acc[s][n][r] + bv;
                if (do_relu) v = fmaxf(v, 0.f);
                C[(size_t)grow * N + gcol] = v;
            }
        }
}

// ---------------------------------------------------------------------------
// Hypergraph scatter pipeline (bandwidth-trivial vs the GEMMs)
// ---------------------------------------------------------------------------
__global__ __launch_bounds__(256)
void zero_kernel(float* __restrict__ p, int n)
{
    int i = blockIdx.x * blockDim.x + threadIdx.x;
    if (i < n) p[i] = 0.f;
}

__global__ __launch_bounds__(256)
void degree_kernel(const int* __restrict__ node, const int* __restrict__ edge,
                   float* __restrict__ D, float* __restrict__ Bd, int E)
{
    int e = blockIdx.x * blockDim.x + threadIdx.x;
    if (e < E) {
        atomicAdd(&D[node[e]], 1.f);
        atomicAdd(&Bd[edge[e]], 1.f);
    }
}

__global__ __launch_bounds__(256)
void scatter64_kernel(const int* __restrict__ idx_dst, const int* __restrict__ idx_src,
                      const float* __restrict__ src, float* __restrict__ dst, int total)
{
    int g = blockIdx.x * blockDim.x + threadIdx.x;
    if (g < total) {
        int e = g >> 6;
        int c = g & 63;
        atomicAdd(&dst[(size_t)idx_dst[e] * CODE + c], src[(size_t)idx_src[e] * CODE + c]);
    }
}

__global__ __launch_bounds__(256)
void scale_rows_kernel(float* __restrict__ m, const float* __restrict__ deg, int total)
{
    int g = blockIdx.x * blockDim.x + threadIdx.x;
    if (g < total) {
        float d = deg[g >> 6];
        m[g] *= (d > 0.f) ? (1.f / d) : 0.f;
    }
}

__global__ __launch_bounds__(256)
void finalize_kernel(const float* __restrict__ hacc, const float* __restrict__ D,
                     const float* __restrict__ bias, float* __restrict__ hid,
                     float* __restrict__ code, int total)
{
    int g = blockIdx.x * blockDim.x + threadIdx.x;
    if (g < total) {
        int r = g >> 6;
        int c = g & 63;
        float d = D[r];
        float v = hacc[g] * ((d > 0.f) ? (1.f / d) : 0.f) + bias[c];
        hid[g]  = v;
        code[g] = tanhf(v);
    }
}

// ---------------------------------------------------------------------------
extern "C" void kernel_launch(void* const* d_in, const int* in_sizes, int n_in,
                              void* d_out, int out_size, void* d_ws, size_t ws_size,
                              hipStream_t stream)
{
    (void)in_sizes; (void)n_in; (void)out_size; (void)ws_size;

    const float* x     = (const float*)d_in[0];   // [N, FEAT]
    const float* W1    = (const float*)d_in[1];   // [FEAT, HID]
    const float* b1    = (const float*)d_in[2];   // [HID]
    const float* theta = (const float*)d_in[3];   // [HID, CODE]
    const float* bias  = (const float*)d_in[4];   // [CODE]
    const int*   hei   = (const int*)d_in[5];     // [2, E]
    const int*   node  = hei;
    const int*   edge  = hei + EE;

    float* feat = (float*)d_out;                  // [N, HID]
    float* hid  = feat + (size_t)NN * HID;        // [N, CODE]
    float* code = hid  + (size_t)NN * CODE;       // [N, CODE]

    float* ws   = (float*)d_ws;
    float* xt   = ws;                             // [N, CODE]
    float* m    = xt   + (size_t)NN * CODE;       // [N, CODE]
    float* hacc = m    + (size_t)NN * CODE;       // [N, CODE]
    float* Dd   = hacc + (size_t)NN * CODE;       // [N]
    float* Bd   = Dd + NN;                        // [N]

    // fc1 + relu : feat = relu(x @ W1 + b1)
    dim3 g1(NN / TM, HID / TN);
    gemm_f32_wmma<<<g1, 256, 0, stream>>>(x, W1, b1, feat, NN, FEAT, HID, 1);

    // xt = feat @ theta
    dim3 g2(NN / TM, CODE / TN);
    gemm_f32_wmma<<<g2, 256, 0, stream>>>(feat, theta, nullptr, xt, NN, HID, CODE, 0);

    // zero accumulators: m, hacc, D, B (contiguous after xt)
    int zcount = NN * CODE * 2 + NN * 2;
    zero_kernel<<<(zcount + 255) / 256, 256, 0, stream>>>(m, zcount);

    degree_kernel<<<(EE + 255) / 256, 256, 0, stream>>>(node, edge, Dd, Bd, EE);

    int tot = EE * CODE;
    scatter64_kernel<<<(tot + 255) / 256, 256, 0, stream>>>(edge, node, xt, m, tot);
    scale_rows_kernel<<<(NN * CODE + 255) / 256, 256, 0, stream>>>(m, Bd, NN * CODE);
    scatter64_kernel<<<(tot + 255) / 256, 256, 0, stream>>>(node, edge, m, hacc, tot);
    finalize_kernel<<<(NN * CODE + 255) / 256, 256, 0, stream>>>(hacc, Dd, bias, hid, code, NN * CODE);
}